// TaylorEBM_50525995270529
// MI455X (gfx1250) — compile-verified
//
#include <hip/hip_runtime.h>

// MI455X / gfx1250 implementation.
//
// Work: 8192 samples x 20 Adam steps x 2 MLP fwd+bwd evals (3->32->32->32->1
// ReLU net) ~= 3 GFLOP; NN search |c_i-c_j| ~= 64M cmps. Tiny, latency-bound:
// the goal is minimal instructions per sample-step. The 32x32 layers are done
// with V_WMMA_F32_16X16X4_F32 (exact f32) batching 16 samples per wave:
//   out[hid, s] : A = W (M=hid, 2 tiles of 16), B = act (K=in, N=sample 16)
// 16 WMMA per layer vs ~512 v_fmac scalar => ~10x fewer instructions.
// Weight A-fragments (w1, w2, w1^T, w2^T) are loop-invariant and hoisted into
// 128 VGPRs before the 20-step loop; activations bounce through sample-major
// LDS buffers with row stride 34 (even -> b64-aligned fragment loads; 34*s
// mod 64 is a permutation over s=0..15 -> conflict-free).
//
// Autodiff note: grad_y of (E + dE) from jvp has zero contribution from dE
// (relu JVP is select(z>0, g, 0); boolean predicate carries no gradient and
// the tangent values are y-independent), so gy = dE/dy at (x,y,c0) and
// gc = dE/dc at (x,y,cc): two plain fwd+bwd passes per step. E and b3 unused.

typedef float v2f __attribute__((ext_vector_type(2)));
typedef float v8f __attribute__((ext_vector_type(8)));

#define NSAMP 16     // samples per wave
#define STR   34     // padded LDS row stride (floats): even + bank-permuting
#define ADAM_STEPS 20

static __device__ __forceinline__ v8f wmma4(v2f a, v2f b, v8f c) {
  // V_WMMA_F32_16X16X4_F32: D = A(16x4) x B(4x16) + C(16x16), all f32.
  return __builtin_amdgcn_wmma_f32_16x16x4_f32(
      /*neg_a=*/false, a, /*neg_b=*/false, b,
      /*c_mod=*/(short)0, c, /*reuse_a=*/false, /*reuse_b=*/false);
}

// ---------------------------------------------------------------------------
// Kernel 1: nearest (non-self) neighbor in 1-D context space.
// dist = |c_i - c_j|; argmin with first-occurrence tie-breaking (strict <,
// ascending scan) matches jnp.argmin. Writes c[argmin] (the c0 value).
// ---------------------------------------------------------------------------
__global__ __launch_bounds__(256)
void nn_kernel(const float* __restrict__ c, float* __restrict__ c0out, int n) {
  __shared__ float sc[8192];
  for (int j = threadIdx.x; j < n; j += blockDim.x) sc[j] = c[j];
  __syncthreads();
  int i = blockIdx.x * blockDim.x + threadIdx.x;
  if (i >= n) return;
  float ci = sc[i];
  float best = 3.4e38f;
  int bj = 0;
  #pragma unroll 4
  for (int j = 0; j < n; ++j) {
    float d = __builtin_fabsf(sc[j] - ci);
    bool better = (d < best) && (j != i);
    best = better ? d : best;
    bj = better ? j : bj;
  }
  c0out[i] = sc[bj];
}

// ---------------------------------------------------------------------------
// Kernel 2: 20-step Adam inner solve. One wave (32 threads) per 16 samples.
// Lanes 0-15 own Adam state for y (sample = lane), lanes 16-31 for c.
// ---------------------------------------------------------------------------
__global__ __launch_bounds__(32)
void solve_kernel(const float* __restrict__ x,
                  const float* __restrict__ c0g,
                  const float* __restrict__ w0g,
                  const float* __restrict__ b0g,
                  const float* __restrict__ w1g,
                  const float* __restrict__ b1g,
                  const float* __restrict__ w2g,
                  const float* __restrict__ b2g,
                  const float* __restrict__ w3g,
                  const float* __restrict__ ymean,
                  float* __restrict__ out, int n) {
  __shared__ float sW1[32 * STR], sW2[32 * STR];    // forward weights
  __shared__ float sW1T[32 * STR], sW2T[32 * STR];  // transposed (backward)
  __shared__ float sW0[96], sB0[32], sB1[32], sB2[32], sW3[32];
  __shared__ float sZ0[16 * STR], sZ1[16 * STR], sZ2[16 * STR]; // pre-acts z
  __shared__ float sG[16 * STR];                                // bwd bounce
  __shared__ float sX[16], sY[16], sC[16], sC0[16];
  __shared__ float sP[32], sGY[16], sGC[16];

  const int lane  = threadIdx.x;      // 0..31, wave32
  const int l16   = lane & 15;        // sample index within wave
  const int hi    = lane >> 4;        // lane half
  const int koff  = hi * 2;           // K sub-offset for A/B fragments
  const int gbase = blockIdx.x * NSAMP;

  // ---- block init: stage weights (+transposes) and sample scalars in LDS ---
  #pragma unroll 4
  for (int k = 0; k < 32; ++k) {
    float v1 = w1g[lane * 32 + k];
    float v2 = w2g[lane * 32 + k];
    sW1[lane * STR + k] = v1;  sW1T[k * STR + lane] = v1;
    sW2[lane * STR + k] = v2;  sW2T[k * STR + lane] = v2;
  }
  for (int q = lane; q < 96; q += 32) sW0[q] = w0g[q];
  sB0[lane] = b0g[lane];
  sB1[lane] = b1g[lane];
  sB2[lane] = b2g[lane];
  sW3[lane] = w3g[lane];
  if (hi == 0) {
    sX[l16]  = x[gbase + l16];
    sY[l16]  = ymean[0];
    sC[l16]  = 0.0f;                 // c_init = zeros_like(c)
    sC0[l16] = c0g[gbase + l16];
  }
  __syncthreads();

  // ---- hoist loop-invariant weight A-fragments into registers -------------
  // [tile][j]: A frag for M-rows 16*tile+l16, K = 4j + 2*hi + {0,1}.
  v2f aW1[2][8], aW2[2][8], aW1T[2][8], aW2T[2][8];
  #pragma unroll
  for (int j = 0; j < 8; ++j) {
    int kb = 4 * j + koff;
    int r0 = l16 * STR + kb, r1 = (16 + l16) * STR + kb;
    aW1[0][j].x  = sW1[r0];   aW1[0][j].y  = sW1[r0 + 1];
    aW1[1][j].x  = sW1[r1];   aW1[1][j].y  = sW1[r1 + 1];
    aW2[0][j].x  = sW2[r0];   aW2[0][j].y  = sW2[r0 + 1];
    aW2[1][j].x  = sW2[r1];   aW2[1][j].y  = sW2[r1 + 1];
    aW1T[0][j].x = sW1T[r0];  aW1T[0][j].y = sW1T[r0 + 1];
    aW1T[1][j].x = sW1T[r1];  aW1T[1][j].y = sW1T[r1 + 1];
    aW2T[0][j].x = sW2T[r0];  aW2T[0][j].y = sW2T[r0 + 1];
    aW2T[1][j].x = sW2T[r1];  aW2T[1][j].y = sW2T[r1 + 1];
  }

  // ---- WMMA layer: z_out[h,s] = W @ relu(z_in)[:,s] + bias -----------------
  auto layer_fwd = [&](const v2f (&Af)[2][8], const float* bias,
                       const float* zin, float* zout) {
    v8f acc0 = {0.f,0.f,0.f,0.f,0.f,0.f,0.f,0.f};
    v8f acc1 = {0.f,0.f,0.f,0.f,0.f,0.f,0.f,0.f};
    #pragma unroll
    for (int j = 0; j < 8; ++j) {
      int kb = 4 * j + koff;
      v2f bf;                         // B frag: relu'd input, sample = l16
      bf.x = __builtin_fmaxf(zin[l16 * STR + kb], 0.f);
      bf.y = __builtin_fmaxf(zin[l16 * STR + kb + 1], 0.f);
      acc0 = wmma4(Af[0][j], bf, acc0);
      acc1 = wmma4(Af[1][j], bf, acc1);
    }
    int hb = 8 * hi;                  // C layout: lane half -> M offset 8
    #pragma unroll
    for (int r = 0; r < 8; ++r) {
      zout[l16 * STR + hb + r]      = acc0[r] + bias[hb + r];
      zout[l16 * STR + 16 + hb + r] = acc1[r] + bias[16 + hb + r];
    }
  };

  // ---- WMMA backward layer: sG <- mask(z) * (W^T @ sG) ---------------------
  auto layer_bwd = [&](const v2f (&Af)[2][8], const float* zmask) {
    v8f acc0 = {0.f,0.f,0.f,0.f,0.f,0.f,0.f,0.f};
    v8f acc1 = {0.f,0.f,0.f,0.f,0.f,0.f,0.f,0.f};
    #pragma unroll
    for (int j = 0; j < 8; ++j) {
      int kb = 4 * j + koff;
      v2f bf;
      bf.x = sG[l16 * STR + kb];
      bf.y = sG[l16 * STR + kb + 1];
      acc0 = wmma4(Af[0][j], bf, acc0);
      acc1 = wmma4(Af[1][j], bf, acc1);
    }
    __syncthreads();                  // all reads of sG done before overwrite
    int hb = 8 * hi;
    #pragma unroll
    for (int r = 0; r < 8; ++r) {
      int h0 = hb + r, h1 = 16 + hb + r;
      sG[l16 * STR + h0] = (zmask[l16 * STR + h0] > 0.f) ? acc0[r] : 0.f;
      sG[l16 * STR + h1] = (zmask[l16 * STR + h1] > 0.f) ? acc1[r] : 0.f;
    }
  };

  // ---- full fwd+bwd eval -> input gradient column (1 = y, 2 = c) -----------
  auto eval_grad = [&](const float* cin, int col, float* gout) {
    float xv = sX[l16], yv = sY[l16], cv = cin[l16];
    #pragma unroll
    for (int i = 0; i < 16; ++i) {    // layer 0: K=3, cheaper scalar
      int h = 16 * hi + i;
      sZ0[l16 * STR + h] =
          sW0[h * 3 + 0] * xv + sW0[h * 3 + 1] * yv + sW0[h * 3 + 2] * cv
          + sB0[h];
    }
    __syncthreads();
    layer_fwd(aW1, sB1, sZ0, sZ1);  __syncthreads();
    layer_fwd(aW2, sB2, sZ1, sZ2);  __syncthreads();
    #pragma unroll
    for (int i = 0; i < 16; ++i) {    // dz2 = mask(z2) * w3
      int h = 16 * hi + i;
      sG[l16 * STR + h] = (sZ2[l16 * STR + h] > 0.f) ? sW3[h] : 0.f;
    }
    __syncthreads();
    layer_bwd(aW2T, sZ1);  __syncthreads();   // dz1
    layer_bwd(aW1T, sZ0);  __syncthreads();   // dz0
    float p = 0.f;                    // dot(w0[:,col], dz0[:,s]) split 2 lanes
    #pragma unroll
    for (int i = 0; i < 16; ++i) {
      int h = 16 * hi + i;
      p += sW0[h * 3 + col] * sG[l16 * STR + h];
    }
    sP[lane] = p;
    __syncthreads();
    if (hi == 0) gout[l16] = sP[l16] + sP[l16 + 16];
    __syncthreads();
  };

  // ---- 20 Adam steps -------------------------------------------------------
  float am = 0.f, av = 0.f;           // this lane's (m, v): y if hi==0 else c
  float b1p = 1.f, b2p = 1.f;
  #pragma unroll 1
  for (int t = 0; t < ADAM_STEPS; ++t) {
    b1p *= 0.9f;
    b2p *= 0.999f;
    eval_grad(sC0, 1, sGY);           // gy = dE/dy at (x, y, c0)
    eval_grad(sC,  2, sGC);           // gc = dE/dc at (x, y, cc)
    float g = (hi == 0) ? sGY[l16] : sGC[l16];
    am = 0.9f * am + 0.1f * g;
    av = 0.999f * av + 0.001f * (g * g);
    float mh = am / (1.f - b1p);
    float vh = av / (1.f - b2p);
    float u = -0.1f * mh / (__builtin_sqrtf(vh) + 1e-8f);
    __syncthreads();
    if (hi == 0) sY[l16] += u; else sC[l16] += u;
    __syncthreads();
  }

  // ---- outputs: [pred (n), c_out (n)] flat ---------------------------------
  if (hi == 0) out[gbase + l16]     = sY[l16];
  else         out[n + gbase + l16] = sC[l16];
}

// ---------------------------------------------------------------------------
extern "C" void kernel_launch(void* const* d_in, const int* in_sizes, int n_in,
                              void* d_out, int out_size, void* d_ws,
                              size_t ws_size, hipStream_t stream) {
  const float* x  = (const float*)d_in[0];
  const float* c  = (const float*)d_in[1];
  const float* w0 = (const float*)d_in[2];
  const float* b0 = (const float*)d_in[3];
  const float* w1 = (const float*)d_in[4];
  const float* b1 = (const float*)d_in[5];
  const float* w2 = (const float*)d_in[6];
  const float* b2 = (const float*)d_in[7];
  const float* w3 = (const float*)d_in[8];
  // d_in[9] = b3 (unused: output E value is never needed, only gradients)
  const float* ym = (const float*)d_in[10];

  const int n = in_sizes[0];          // 8192
  float* c0ws = (float*)d_ws;         // n floats of scratch: c0_batch values
  float* outf = (float*)d_out;

  nn_kernel<<<(n + 255) / 256, 256, 0, stream>>>(c, c0ws, n);
  solve_kernel<<<n / NSAMP, 32, 0, stream>>>(x, c0ws, w0, b0, w1, b1, w2, b2,
                                             w3, ym, outf, n);
}